// spikeLN_77360950935786
// MI455X (gfx1250) — compile-verified
//
#include <hip/hip_runtime.h>
#include <cstdint>
#include <cstddef>

// spikeLN for MI455X (gfx1250): OATN spike-quantize + LayerNorm(D=4096) + affine.
// Memory-bound (512 MB traffic, ~22 us @ 23.3 TB/s). Persistent workgroups with
// double-buffered async global->LDS row staging (ASYNCcnt pipeline), NT hints on
// the one-shot 256MB input stream (> 192MB L2), wave32 reductions,
// register-resident quantized values, non-temporal stores.

#define DIM       4096
#define NTHREADS  256
#define CHUNKS    4      // 4 x float4 per thread = 16 elements
#define MAX_WG    2048

typedef float v4f __attribute__((ext_vector_type(4)));

__device__ __forceinline__ float oatn_q(float x) {
    // clamp to [-500, 500]
    float xc = fminf(fmaxf(x, -500.0f), 500.0f);
    float a  = fabsf(xc);
    bool  sm = a < 10.0f;                       // |x| < 10 -> v_max = 10 else 500
    // q = floor(a / v_max * 65536) / 65536 * v_max
    float q  = floorf(a * (sm ? 6553.6f : 131.072f))
             * (sm ? 1.52587890625e-4f : 7.62939453125e-3f);  // v_max/65536
    // saturate at v_max * (1 - 2^-16)
    q = fminf(q, sm ? 9.999847412109375f : 499.99237060546875f);
    return copysignf(q, xc);                    // restore sign (q==0 for x==0)
}

__device__ __forceinline__ void stage_row_async(unsigned lds_byte, const float* gp) {
    // Each instruction: 256 lanes x 16B = 4KB contiguous. Same IOFFSET is added
    // to BOTH the LDS and global addresses (ISA 15.18 async pseudocode), so one
    // VGPR base each side covers the whole 16KB row in 4 issues. TH_LOAD_NT:
    // the input is streamed exactly once (256MB > 192MB L2) - don't let it
    // evict the resident weight/bias + store write-combining lines.
    asm volatile("global_load_async_to_lds_b128 %0, %1, off th:TH_LOAD_NT"               :: "v"(lds_byte), "v"(gp) : "memory");
    asm volatile("global_load_async_to_lds_b128 %0, %1, off offset:4096 th:TH_LOAD_NT"   :: "v"(lds_byte), "v"(gp) : "memory");
    asm volatile("global_load_async_to_lds_b128 %0, %1, off offset:8192 th:TH_LOAD_NT"   :: "v"(lds_byte), "v"(gp) : "memory");
    asm volatile("global_load_async_to_lds_b128 %0, %1, off offset:12288 th:TH_LOAD_NT"  :: "v"(lds_byte), "v"(gp) : "memory");
}

__global__ __launch_bounds__(NTHREADS)
void spikeln_kernel(const float* __restrict__ x,
                    const float* __restrict__ w,
                    const float* __restrict__ b,
                    float* __restrict__ out,
                    int rows) {
    __shared__ __align__(16) float buf[2][DIM];   // double-buffered row staging
    __shared__ float red[2][8];                   // cross-wave partials (8 waves)

    const int tid    = threadIdx.x;
    const int stride = gridDim.x;

    // LDS byte addresses for this thread's 16B slot in each buffer
    const unsigned lds0 = (unsigned)(uintptr_t)(&buf[0][0]) + (unsigned)tid * 16u;
    const unsigned lds1 = (unsigned)(uintptr_t)(&buf[1][0]) + (unsigned)tid * 16u;

    int r = blockIdx.x;
    const size_t rowstep = (size_t)stride * DIM;

    // Incremental pointers (no per-iteration 64-bit multiplies)
    const float* gnext = x + (size_t)r * DIM + (size_t)tid * 4;   // staging source
    float*       op    = out + (size_t)r * DIM;                   // output row base

    // Prologue: stage first row into buffer 0
    stage_row_async(lds0, gnext);
    gnext += rowstep;

    int parity = 0;
    for (; r < rows; r += stride) {
        const int rn = r + stride;                // uniform across the block
        if (rn < rows) {
            // Kick off next row into the other buffer, then wait only for the
            // 4 oldest (current-row) copies: ASYNCcnt <= 4.
            stage_row_async(parity ? lds0 : lds1, gnext);
            gnext += rowstep;
            asm volatile("s_wait_asynccnt 0x4" ::: "memory");
        } else {
            asm volatile("s_wait_asynccnt 0x0" ::: "memory");
        }
        __syncthreads();   // all waves' staging of buf[parity] is now visible

        // ---- pass 1: quantize + accumulate (values stay in VGPRs) ----
        const v4f* lb = reinterpret_cast<const v4f*>(&buf[parity][0]);
        float s[CHUNKS][4];
        float sum = 0.0f, ss = 0.0f;
#pragma unroll
        for (int c = 0; c < CHUNKS; ++c) {
            v4f v = lb[c * NTHREADS + tid];       // ds_load_b128, conflict-free
#pragma unroll
            for (int j = 0; j < 4; ++j) {
                float q = oatn_q(v[j]);
                s[c][j] = q;
                sum += q;
                ss  += q * q;
            }
        }
        // wave32 butterfly reduction
#pragma unroll
        for (int off = 16; off >= 1; off >>= 1) {
            sum += __shfl_xor(sum, off, 32);
            ss  += __shfl_xor(ss,  off, 32);
        }
        const int wid = tid >> 5;
        if ((tid & 31) == 0) { red[0][wid] = sum; red[1][wid] = ss; }
        __syncthreads();
        sum = 0.0f; ss = 0.0f;
#pragma unroll
        for (int i = 0; i < 8; ++i) { sum += red[0][i]; ss += red[1][i]; }

        const float mu   = sum * (1.0f / DIM);
        const float var  = fmaxf(ss * (1.0f / DIM) - mu * mu, 0.0f);
        const float rinv = rsqrtf(var + 1e-5f);

        // Last LDS access (buf + red) is above: release the buffers HERE so the
        // store pass below overlaps with other waves starting the next row.
        __syncthreads();

        // ---- pass 2: normalize + affine, non-temporal streaming stores ----
        const v4f* w4 = reinterpret_cast<const v4f*>(w);
        const v4f* b4 = reinterpret_cast<const v4f*>(b);
        v4f* o4 = reinterpret_cast<v4f*>(op);
#pragma unroll
        for (int c = 0; c < CHUNKS; ++c) {
            const int idx = c * NTHREADS + tid;   // float4 index within the row
            v4f wv = w4[idx];
            v4f bv = b4[idx];
            v4f ov;
#pragma unroll
            for (int j = 0; j < 4; ++j)
                ov[j] = (s[c][j] - mu) * rinv * wv[j] + bv[j];
            __builtin_nontemporal_store(ov, &o4[idx]);
        }
        op += rowstep;
        parity ^= 1;
    }
}

extern "C" void kernel_launch(void* const* d_in, const int* in_sizes, int n_in,
                              void* d_out, int out_size, void* d_ws, size_t ws_size,
                              hipStream_t stream) {
    const float* x = (const float*)d_in[0];
    const float* w = (const float*)d_in[1];
    const float* b = (const float*)d_in[2];
    float* out = (float*)d_out;

    const int d    = in_sizes[1];                 // 4096
    const int rows = in_sizes[0] / d;             // 16384
    const int nwg  = rows < MAX_WG ? rows : MAX_WG;

    spikeln_kernel<<<nwg, NTHREADS, 0, stream>>>(x, w, b, out, rows);
}